// MultiScaleDogFilter_3891240370766
// MI455X (gfx1250) — compile-verified
//
#include <hip/hip_runtime.h>

// Multi-scale DoG on gfx1250: separable Gaussian = banded GEMM on the
// FP32 matrix pipe (v_wmma_f32_16x16x4_f32). H-pass stages reflect-padded
// rows in LDS; V-pass stages padded column strips and fuses the subband
// difference. Output slot 0 of d_out doubles as the H-blur scratch.
//
// Band weights live in a ZERO-PADDED LDS array (gsPad[u+15] = g[u], else 0)
// so fragment builds are unconditional LDS reads: no divergent saveexec
// regions inside the WMMA loop.

typedef __attribute__((ext_vector_type(2))) float v2f;
typedef __attribute__((ext_vector_type(8))) float v8f;

#define PLANE_SZ (512 * 512)
#define HP_STRIDE 645   // odd stride: A-frag reads (16 rows, same col) hit 16 distinct banks
#define GPAD 15         // zero-pad offset for band weights
#define GSZ 160         // >= K4max-1 + GPAD + 2 = 131+15+2

__device__ __forceinline__ int reflect512(int x) {
  x = (x < 0) ? -x : x;
  return (x > 511) ? (1022 - x) : x;
}

// Zero-padded normalized Gaussian into gs[GPAD .. GPAD+ks).
__device__ __forceinline__ void compute_gauss(float* gs, float sigma, int ks) {
  for (int i = threadIdx.x; i < GSZ; i += 512) gs[i] = 0.0f;
  __syncthreads();
  if (threadIdx.x == 0) {
    const int p = ks >> 1;
    const float inv2s2 = 1.0f / (2.0f * sigma * sigma);
    float sum = 0.0f;
    for (int t = 0; t < ks; ++t) {
      float d = (float)(t - p);
      float e = __expf(-d * d * inv2s2);
      gs[GPAD + t] = e;
      sum += e;
    }
    const float inv = 1.0f / sum;
    for (int t = 0; t < ks; ++t) gs[GPAD + t] *= inv;
  }
  __syncthreads();
}

// ---------------------------------------------------------------------------
// Horizontal (along W) Gaussian pass:  out_slot0 = img (*) g  (row conv)
// Tile GEMM: Out(16x16) = A(16xK) * Band(Kx16), Band[k][n] = g[k-n].
// grid.x = 24 planes * 32 row-strips; block = 512 threads (16 waves).
// ---------------------------------------------------------------------------
__global__ __launch_bounds__(512) void dog_hpass(const float* __restrict__ img,
                                                 float* __restrict__ out,
                                                 float sigma, int ks, int K4) {
  __shared__ float gs[GSZ];
  __shared__ float pad[16 * HP_STRIDE];

  const int plane = blockIdx.x >> 5;      // 0..23  (= b*3 + c)
  const int rt    = blockIdx.x & 31;      // row-strip index
  const int bIdx  = plane / 3;
  const int cIdx  = plane % 3;
  const int r0    = rt * 16;
  const int p     = ks >> 1;
  const int padW  = 512 + K4;

  compute_gauss(gs, sigma, ks);

  // Stage 16 reflect-padded rows into LDS.
  {
    const int r = threadIdx.x >> 5;                 // 0..15
    const float* srow = img + (size_t)plane * PLANE_SZ + (size_t)(r0 + r) * 512;
    for (int x = threadIdx.x & 31; x < padW; x += 32)
      pad[r * HP_STRIDE + x] = srow[reflect512(x - p)];
  }
  __syncthreads();

  const int lane   = threadIdx.x & 31;
  const int wv     = threadIdx.x >> 5;              // 0..15
  const int m      = lane & 15;                     // row (A) / col (B,D) within tile
  const int koff   = (lane >> 4) << 1;              // 0 or 2
  const int nsteps = K4 >> 2;
  const int c0     = wv * 32;                       // two adjacent 16-col tiles per wave

  // Unconditional base pointers for fragment builds.
  const float* bandBase = gs + (GPAD + koff - m);   // band[kb=4j]   = bandBase[4j]
  const float* aBase0   = pad + (m * HP_STRIDE + c0 + koff);
  const float* aBase1   = aBase0 + 16;

  v8f acc0 = {0.f, 0.f, 0.f, 0.f, 0.f, 0.f, 0.f, 0.f};
  v8f acc1 = {0.f, 0.f, 0.f, 0.f, 0.f, 0.f, 0.f, 0.f};

  for (int j = 0; j < nsteps; ++j) {
    const int k4 = j << 2;
    v2f bm, a0, a1;
    bm.x = bandBase[k4];
    bm.y = bandBase[k4 + 1];
    a0.x = aBase0[k4];
    a0.y = aBase0[k4 + 1];
    a1.x = aBase1[k4];
    a1.y = aBase1[k4 + 1];
    acc0 = __builtin_amdgcn_wmma_f32_16x16x4_f32(false, a0, false, bm, (short)0,
                                                 acc0, false, false);
    acc1 = __builtin_amdgcn_wmma_f32_16x16x4_f32(false, a1, false, bm, (short)0,
                                                 acc1, false, false);
  }

  // Store to out slot 0 (scratch). D layout: d[v] = D[v + 8*(lane/16)][lane%16].
  float* dst = out + ((size_t)(bIdx * 7 + 0) * 3 + cIdx) * PLANE_SZ;
  const int rowb = r0 + ((lane >> 4) << 3);
  float* d0 = dst + (size_t)rowb * 512 + c0 + m;
  float* d1 = d0 + 16;
#pragma unroll
  for (int v = 0; v < 8; ++v) {
    d0[(size_t)v * 512] = acc0[v];
    d1[(size_t)v * 512] = acc1[v];
  }
}

// ---------------------------------------------------------------------------
// Vertical (along H) pass + fused DoG difference.
// Tile GEMM: Out(16x16) = Band(16xK) * A(Kx16), Band[m][k] = g[k-m].
// Reads H-blur from slot 0; writes blur_s -> slot s+1, blur_s - prev -> slot s.
// prev = img for s==0, else slot s (blur_{s-1}).
// grid.x = 24 planes * 32 col-strips; block = 512 threads.
// ---------------------------------------------------------------------------
__global__ __launch_bounds__(512) void dog_vpass(const float* __restrict__ img,
                                                 float* __restrict__ out,
                                                 int s, float sigma, int ks, int K4) {
  __shared__ float gs[GSZ];
  __shared__ float pad[644 * 16];

  const int plane = blockIdx.x >> 5;
  const int ct    = blockIdx.x & 31;
  const int bIdx  = plane / 3;
  const int cIdx  = plane % 3;
  const int c0    = ct * 16;
  const int p     = ks >> 1;
  const int padH  = 512 + K4;

  compute_gauss(gs, sigma, ks);

  // Stage reflect-padded column strip (padH rows x 16 cols) from slot 0.
  {
    const float* src = out + ((size_t)(bIdx * 7 + 0) * 3 + cIdx) * PLANE_SZ;
    const int n = threadIdx.x & 15;
    for (int r = threadIdx.x >> 4; r < padH; r += 32)
      pad[r * 16 + n] = src[(size_t)reflect512(r - p) * 512 + c0 + n];
  }
  __syncthreads();

  const int lane   = threadIdx.x & 31;
  const int wv     = threadIdx.x >> 5;
  const int m      = lane & 15;
  const int koff   = (lane >> 4) << 1;
  const int nsteps = K4 >> 2;
  const int r0     = wv * 32;                       // two adjacent 16-row tiles per wave

  const float* bandBase = gs + (GPAD + koff - m);   // band[kb=4j] = bandBase[4j]
  const float* bBase0   = pad + ((r0 + koff) * 16 + m);
  const float* bBase1   = bBase0 + 16 * 16;

  v8f acc0 = {0.f, 0.f, 0.f, 0.f, 0.f, 0.f, 0.f, 0.f};
  v8f acc1 = {0.f, 0.f, 0.f, 0.f, 0.f, 0.f, 0.f, 0.f};

  for (int j = 0; j < nsteps; ++j) {
    const int k4 = j << 2;
    v2f a, b0, b1;
    a.x = bandBase[k4];
    a.y = bandBase[k4 + 1];
    b0.x = bBase0[k4 * 16];
    b0.y = bBase0[k4 * 16 + 16];
    b1.x = bBase1[k4 * 16];
    b1.y = bBase1[k4 * 16 + 16];
    acc0 = __builtin_amdgcn_wmma_f32_16x16x4_f32(false, a, false, b0, (short)0,
                                                 acc0, false, false);
    acc1 = __builtin_amdgcn_wmma_f32_16x16x4_f32(false, a, false, b1, (short)0,
                                                 acc1, false, false);
  }

  const float* prevB = (s == 0)
      ? (img + (size_t)plane * PLANE_SZ)
      : (out + ((size_t)(bIdx * 7 + s) * 3 + cIdx) * PLANE_SZ);
  float* diffB = out + ((size_t)(bIdx * 7 + s) * 3 + cIdx) * PLANE_SZ;
  float* blurB = out + ((size_t)(bIdx * 7 + s + 1) * 3 + cIdx) * PLANE_SZ;

  const int rowb = r0 + ((lane >> 4) << 3);
  const size_t off0 = (size_t)rowb * 512 + c0 + m;
  const size_t off1 = off0 + (size_t)16 * 512;
#pragma unroll
  for (int v = 0; v < 8; ++v) {
    const size_t o0 = off0 + (size_t)v * 512;
    const size_t o1 = off1 + (size_t)v * 512;
    const float bl0 = acc0[v];
    const float bl1 = acc1[v];
    const float pv0 = prevB[o0];
    const float pv1 = prevB[o1];
    blurB[o0] = bl0;
    blurB[o1] = bl1;
    diffB[o0] = bl0 - pv0;
    diffB[o1] = bl1 - pv1;
  }
}

extern "C" void kernel_launch(void* const* d_in, const int* in_sizes, int n_in,
                              void* d_out, int out_size, void* d_ws, size_t ws_size,
                              hipStream_t stream) {
  (void)in_sizes; (void)n_in; (void)out_size; (void)d_ws; (void)ws_size;
  const float* img = (const float*)d_in[0];
  float* out = (float*)d_out;

  static const float SIG[6] = {0.6f, 1.2f, 2.4f, 4.8f, 9.6f, 19.2f};
  static const int   KS[6]  = {7, 9, 17, 31, 59, 117};
  // K4 = ceil((ks + 15) / 4) * 4
  static const int   K4[6]  = {24, 24, 32, 48, 76, 132};

  const dim3 grid(24 * 32);   // planes * 16-wide strips
  const dim3 block(512);      // 16 waves

  for (int s = 0; s < 6; ++s) {
    dog_hpass<<<grid, block, 0, stream>>>(img, out, SIG[s], KS[s], K4[s]);
    dog_vpass<<<grid, block, 0, stream>>>(img, out, s, SIG[s], KS[s], K4[s]);
  }
}